// Seq2SeqDecoder_23931557773552
// MI455X (gfx1250) — compile-verified
//
#include <hip/hip_runtime.h>
#include <math.h>

typedef __attribute__((ext_vector_type(2))) float v2f;
typedef __attribute__((ext_vector_type(8))) float v8f;

#define BB 32          // batch
#define HH 512         // hidden
#define EE 256         // embed
#define VV 32000       // vocab
#define TT 64          // seq len
#define FF 1280        // 2H + E (feat width)
#define GG 2048        // 4H (gates)
#define XK 768         // H + E (x width for W_ih)
#define NLB 250        // logits blocks (argmax partials)

#define PROW 66        // floats per K-pair row in LDS: 2*32 batch + 2 pad

// ---------------------------------------------------------------------------
// init: feat[:,0:512]=z=h0, feat[:,768:1280]=h0, c=c0, y=y0
// ---------------------------------------------------------------------------
__global__ __launch_bounds__(256) void init_kernel(const int* __restrict__ y0,
                                                   const float* __restrict__ h0,
                                                   const float* __restrict__ c0,
                                                   float* __restrict__ feat,
                                                   float* __restrict__ cst,
                                                   int* __restrict__ ycur) {
    int tid = blockIdx.x * 256 + threadIdx.x;        // 0 .. 16383
    int b = tid >> 9;
    int j = tid & 511;
    float h = h0[tid];
    feat[b * FF + j] = h;            // z (constant latent)
    feat[b * FF + XK + j] = h;       // h_prev
    cst[tid] = c0[tid];
    if (tid < BB) ycur[tid] = y0[tid];
}

// ---------------------------------------------------------------------------
// fused: final argmax over logits-block partials (wave 0) + embedding gather.
// grid: 32 blocks (one per batch row) x 256 threads (= E columns).
// first!=0: skip reduce, take y from ycur (set by init).
// ---------------------------------------------------------------------------
__global__ __launch_bounds__(256) void reduce_gather_kernel(const float* __restrict__ pV,
                                                            const int* __restrict__ pI,
                                                            const float* __restrict__ emb,
                                                            float* __restrict__ feat,
                                                            int* __restrict__ ycur,
                                                            int first) {
    __shared__ int ysh;
    const int b = blockIdx.x;
    const int tid = threadIdx.x;
    if (tid < 32) {
        int y;
        if (first) {
            y = ycur[b];
        } else {
            // lane-strided scan of 250 partials; (max, lowest vocab idx) rule
            float bv = -INFINITY;
            int bi = 0x7fffffff;
            for (int g = tid; g < NLB; g += 32) {
                float v = pV[g * BB + b];
                int i = pI[g * BB + b];
                if (v > bv || (v == bv && i < bi)) { bv = v; bi = i; }
            }
            for (int off = 16; off > 0; off >>= 1) {   // wave32 butterfly
                float ov = __shfl_xor(bv, off, 32);
                int oi = __shfl_xor(bi, off, 32);
                if (ov > bv || (ov == bv && oi < bi)) { bv = ov; bi = oi; }
            }
            y = bi;
            if (tid == 0) ycur[b] = y;
        }
        if (tid == 0) ysh = y;
    }
    __syncthreads();
    feat[b * FF + HH + tid] = emb[(size_t)ysh * EE + tid];   // gather emb[y]
}

// ---------------------------------------------------------------------------
// helper: stage a K-chunk of feat into LDS, pair-interleaved:
//   featP[(k/2)*PROW + 2*b + (k&1)] = feat[b][koff + k],  k in [0, kchunk)
// one b64 LDS store per K-pair; global side fully coalesced float4 loads.
// ---------------------------------------------------------------------------
__device__ __forceinline__ void stage_feat(float* __restrict__ featP,
                                           const float* __restrict__ feat,
                                           int koff, int kchunk, int tid) {
    for (int i = tid; i < (BB * kchunk) / 4; i += 256) {
        int base = i * 4;
        int bb = base / kchunk;
        int kk = base % kchunk;
        float4 v = *(const float4*)(feat + bb * FF + koff + kk);
        v2f p0; p0.x = v.x; p0.y = v.y;
        v2f p1; p1.x = v.z; p1.y = v.w;
        int r = (kk >> 1);                       // even (kk multiple of 4)
        *(v2f*)(&featP[r * PROW + 2 * bb]) = p0;
        *(v2f*)(&featP[(r + 1) * PROW + 2 * bb]) = p1;
    }
}

// ---------------------------------------------------------------------------
// gates GEMM via fp32 WMMA: gates(32x2048) = feat[:,0:768]@W_ih^T + feat[:,768:1280]@W_hh^T
// grid: 16 blocks x 256 threads; wave w owns 16 gate rows.
// ---------------------------------------------------------------------------
__global__ __launch_bounds__(256) void gates_kernel(const float* __restrict__ Wih,
                                                    const float* __restrict__ Whh,
                                                    const float* __restrict__ feat,
                                                    float* __restrict__ gates) {
    __shared__ float featP[(256 / 2) * PROW];        // K-chunk, pair-interleaved

    const int tid  = threadIdx.x;
    const int wave = tid >> 5;
    const int lid  = tid & 31;
    const int half = lid >> 4;
    const int l    = lid & 15;
    const int nrow = blockIdx.x * 128 + wave * 16 + l;   // this lane's A row (gate idx)

    const float* ArowX = Wih + (size_t)nrow * XK;
    const float* ArowH = Whh + (size_t)nrow * HH;

    v8f acc0 = {};
    v8f acc1 = {};

    // ---- segment 1: x part (K = 0..768 of feat, W_ih) ----
    for (int kc = 0; kc < XK; kc += 256) {
        stage_feat(featP, feat, kc, 256, tid);
        __syncthreads();
        for (int k0 = 0; k0 < 256; k0 += 4) {
            v2f a = *(const v2f*)(ArowX + kc + k0 + 2 * half);
            int kp = (k0 >> 1) + half;
            v2f b0 = *(const v2f*)(&featP[kp * PROW + 2 * l]);
            v2f b1 = *(const v2f*)(&featP[kp * PROW + 2 * l + 32]);
            acc0 = __builtin_amdgcn_wmma_f32_16x16x4_f32(false, a, false, b0,
                                                         (short)0, acc0, false, false);
            acc1 = __builtin_amdgcn_wmma_f32_16x16x4_f32(false, a, false, b1,
                                                         (short)0, acc1, false, false);
        }
        __syncthreads();
    }

    // ---- segment 2: h part (K = 768..1280 of feat, W_hh) ----
    for (int kc = 0; kc < HH; kc += 256) {
        stage_feat(featP, feat, XK + kc, 256, tid);
        __syncthreads();
        for (int k0 = 0; k0 < 256; k0 += 4) {
            v2f a = *(const v2f*)(ArowH + kc + k0 + 2 * half);
            int kp = (k0 >> 1) + half;
            v2f b0 = *(const v2f*)(&featP[kp * PROW + 2 * l]);
            v2f b1 = *(const v2f*)(&featP[kp * PROW + 2 * l + 32]);
            acc0 = __builtin_amdgcn_wmma_f32_16x16x4_f32(false, a, false, b0,
                                                         (short)0, acc0, false, false);
            acc1 = __builtin_amdgcn_wmma_f32_16x16x4_f32(false, a, false, b1,
                                                         (short)0, acc1, false, false);
        }
        __syncthreads();
    }

    // D layout: lane l holds gates[n = base + r + 8*half][batch = l (acc0) / l+16 (acc1)]
    for (int r = 0; r < 8; ++r) {
        int n = blockIdx.x * 128 + wave * 16 + r + 8 * half;
        gates[(size_t)l * GG + n] = acc0[r];
        gates[(size_t)(l + 16) * GG + n] = acc1[r];
    }
}

// ---------------------------------------------------------------------------
// LSTM pointwise: gate order i,f,g,o (torch); updates c and h (feat cols 768:1280)
// ---------------------------------------------------------------------------
__global__ __launch_bounds__(256) void lstm_kernel(const float* __restrict__ gates,
                                                   const float* __restrict__ b_ih,
                                                   const float* __restrict__ b_hh,
                                                   float* __restrict__ cst,
                                                   float* __restrict__ feat) {
    int tid = blockIdx.x * 256 + threadIdx.x;        // 0 .. 16383
    int b = tid >> 9;
    int j = tid & 511;
    const float* g = gates + (size_t)b * GG;
    float gi = g[j]           + b_ih[j]           + b_hh[j];
    float gf = g[512 + j]     + b_ih[512 + j]     + b_hh[512 + j];
    float gg = g[1024 + j]    + b_ih[1024 + j]    + b_hh[1024 + j];
    float go = g[1536 + j]    + b_ih[1536 + j]    + b_hh[1536 + j];
    float si = 1.0f / (1.0f + expf(-gi));
    float sf = 1.0f / (1.0f + expf(-gf));
    float tg = tanhf(gg);
    float so = 1.0f / (1.0f + expf(-go));
    float cn = sf * cst[tid] + si * tg;
    cst[tid] = cn;
    feat[b * FF + XK + j] = so * tanhf(cn);
}

// ---------------------------------------------------------------------------
// logits GEMM via fp32 WMMA + per-block argmax partials.
// grid: 250 blocks x 256 threads; block owns 128 vocab rows; wave owns 16.
// out layout: (B, V, T) -> out[(b*V + v)*T + t]
// ---------------------------------------------------------------------------
__global__ __launch_bounds__(256) void logits_kernel(const float* __restrict__ Wout,
                                                     const float* __restrict__ bout,
                                                     const float* __restrict__ feat,
                                                     float* __restrict__ out,
                                                     float* __restrict__ pV,
                                                     int* __restrict__ pI,
                                                     int t) {
    __shared__ float featP[(320 / 2) * PROW];        // K-chunk, pair-interleaved
    __shared__ float redV[8 * BB];
    __shared__ int   redI[8 * BB];

    const int tid  = threadIdx.x;
    const int wave = tid >> 5;
    const int lid  = tid & 31;
    const int half = lid >> 4;
    const int l    = lid & 15;
    const int vbase = blockIdx.x * 128 + wave * 16;
    const int vrow  = vbase + l;                     // this lane's A row (vocab idx)
    const float* Arow = Wout + (size_t)vrow * FF;

    v8f acc0 = {};
    v8f acc1 = {};

    for (int kc = 0; kc < FF; kc += 320) {
        stage_feat(featP, feat, kc, 320, tid);
        __syncthreads();
        if (kc + 320 < FF) {
            __builtin_prefetch(Arow + kc + 320 + 2 * half, 0, 1);  // next weight chunk
        }
        for (int k0 = 0; k0 < 320; k0 += 4) {
            v2f a = *(const v2f*)(Arow + kc + k0 + 2 * half);      // streams W_out row
            int kp = (k0 >> 1) + half;
            v2f b0 = *(const v2f*)(&featP[kp * PROW + 2 * l]);
            v2f b1 = *(const v2f*)(&featP[kp * PROW + 2 * l + 32]);
            acc0 = __builtin_amdgcn_wmma_f32_16x16x4_f32(false, a, false, b0,
                                                         (short)0, acc0, false, false);
            acc1 = __builtin_amdgcn_wmma_f32_16x16x4_f32(false, a, false, b1,
                                                         (short)0, acc1, false, false);
        }
        __syncthreads();
    }

    // epilogue: bias, store to (B,V,T), per-lane argmax (first-occurrence tie-break)
    float best0 = -INFINITY, best1 = -INFINITY;
    int bi0 = 0, bi1 = 0;
    for (int r = 0; r < 8; ++r) {
        int v = vbase + r + 8 * half;                // ascending in r
        float bias = bout[v];
        float x0 = acc0[r] + bias;                   // batch l
        float x1 = acc1[r] + bias;                   // batch l+16
        out[((size_t)l * VV + v) * TT + t] = x0;
        out[((size_t)(l + 16) * VV + v) * TT + t] = x1;
        if (x0 > best0) { best0 = x0; bi0 = v; }
        if (x1 > best1) { best1 = x1; bi1 = v; }
    }
    // merge the two lane-halves (rows +0..7 live in half 0, +8..15 in half 1)
    {
        float ov = __shfl_xor(best0, 16, 32);
        int   oi = __shfl_xor(bi0, 16, 32);
        if (ov > best0 || (ov == best0 && oi < bi0)) { best0 = ov; bi0 = oi; }
        ov = __shfl_xor(best1, 16, 32);
        oi = __shfl_xor(bi1, 16, 32);
        if (ov > best1 || (ov == best1 && oi < bi1)) { best1 = ov; bi1 = oi; }
    }
    if (half == 0) {
        redV[wave * BB + l] = best0;       redI[wave * BB + l] = bi0;
        redV[wave * BB + 16 + l] = best1;  redI[wave * BB + 16 + l] = bi1;
    }
    __syncthreads();
    if (tid < BB) {                        // reduce 8 waves (ascending vocab order)
        float bv = redV[tid];
        int bi = redI[tid];
        for (int w = 1; w < 8; ++w) {
            float v2 = redV[w * BB + tid];
            int i2 = redI[w * BB + tid];
            if (v2 > bv || (v2 == bv && i2 < bi)) { bv = v2; bi = i2; }
        }
        pV[blockIdx.x * BB + tid] = bv;
        pI[blockIdx.x * BB + tid] = bi;
    }
}

// ---------------------------------------------------------------------------
extern "C" void kernel_launch(void* const* d_in, const int* in_sizes, int n_in,
                              void* d_out, int out_size, void* d_ws, size_t ws_size,
                              hipStream_t stream) {
    const int*   y0   = (const int*)d_in[0];
    const float* h0   = (const float*)d_in[1];
    const float* c0   = (const float*)d_in[2];
    const float* emb  = (const float*)d_in[3];
    const float* Wih  = (const float*)d_in[4];
    const float* Whh  = (const float*)d_in[5];
    const float* bih  = (const float*)d_in[6];
    const float* bhh  = (const float*)d_in[7];
    const float* Wout = (const float*)d_in[8];
    const float* bout = (const float*)d_in[9];
    float* out = (float*)d_out;

    char* ws = (char*)d_ws;
    float* feat  = (float*)(ws + 0);          // 32*1280*4   = 163840
    float* cst   = (float*)(ws + 163840);     // 32*512*4    =  65536
    float* gates = (float*)(ws + 229376);     // 32*2048*4   = 262144
    int*   ycur  = (int*)(ws + 491520);       // 32*4        =    128
    float* pV    = (float*)(ws + 491648);     // 250*32*4    =  32000
    int*   pI    = (int*)(ws + 523648);       // 250*32*4    =  32000

    init_kernel<<<64, 256, 0, stream>>>(y0, h0, c0, feat, cst, ycur);
    for (int t = 0; t < TT; ++t) {
        reduce_gather_kernel<<<32, 256, 0, stream>>>(pV, pI, emb, feat, ycur, t == 0);
        gates_kernel<<<16, 256, 0, stream>>>(Wih, Whh, feat, gates);
        lstm_kernel<<<64, 256, 0, stream>>>(gates, bih, bhh, cst, feat);
        logits_kernel<<<250, 256, 0, stream>>>(Wout, bout, feat, out, pV, pI, t);
    }
}